// AttnLatentScaleSelectionHead_28518582845718
// MI455X (gfx1250) — compile-verified
//
#include <hip/hip_runtime.h>
#include <stdint.h>

#define NN     1125
#define NNSQ   (1125*1125)
#define NMAT   8          // blocks 8..11 x batch 0..1 ; global matrix index = 16 + m
#define RANK0  126561u    // k-1, k = int(1125*1125*0.1) = 126562
#define QS     1128       // padded Q row stride (floats), covers ks+1 <= 1127
#define NTILES 71         // ceil(1125/16)

typedef __attribute__((ext_vector_type(2))) float v2f;
typedef __attribute__((ext_vector_type(8))) float v8f;

// ---------------- radix-select (exact k-th smallest, 3 passes) ----------------

__global__ void k_zero(uint32_t* __restrict__ hist, uint32_t* __restrict__ state, int init_state) {
  int i = blockIdx.x * blockDim.x + threadIdx.x;
  for (int j = i; j < NMAT * 2048; j += gridDim.x * blockDim.x) hist[j] = 0;
  if (init_state && i < NMAT) { state[2*i] = 0u; state[2*i+1] = RANK0; }
}

__global__ void k_hist(const float* __restrict__ attns, uint32_t* __restrict__ hist,
                       const uint32_t* __restrict__ state,
                       int plen, int shiftbits, int nbins) {
  __shared__ uint32_t h[2048];
  int m = blockIdx.y;
  for (int i = threadIdx.x; i < nbins; i += blockDim.x) h[i] = 0u;
  __syncthreads();
  const float* base = attns + (size_t)(16 + m) * NNSQ;   // NNSQ odd: not 16B aligned for every m
  const float4* base4 = (const float4*)base;
  uint32_t prefix = state[2*m];
  uint32_t mask = (uint32_t)(nbins - 1);
  int stride = gridDim.x * blockDim.x;
  const int N4 = NNSQ / 4;                               // 316406, 1 scalar tail elem
  for (int i = blockIdx.x * blockDim.x + threadIdx.x; i < N4; i += stride) {
    float4 v = base4[i];
    uint32_t b0 = __float_as_uint(v.x), b1 = __float_as_uint(v.y);
    uint32_t b2 = __float_as_uint(v.z), b3 = __float_as_uint(v.w);
    if (plen == 0 || (b0 >> (32 - plen)) == prefix) atomicAdd(&h[(b0 >> shiftbits) & mask], 1u);
    if (plen == 0 || (b1 >> (32 - plen)) == prefix) atomicAdd(&h[(b1 >> shiftbits) & mask], 1u);
    if (plen == 0 || (b2 >> (32 - plen)) == prefix) atomicAdd(&h[(b2 >> shiftbits) & mask], 1u);
    if (plen == 0 || (b3 >> (32 - plen)) == prefix) atomicAdd(&h[(b3 >> shiftbits) & mask], 1u);
  }
  if (blockIdx.x == 0 && threadIdx.x == 0) {
    uint32_t bl = __float_as_uint(base[NNSQ - 1]);
    if (plen == 0 || (bl >> (32 - plen)) == prefix) atomicAdd(&h[(bl >> shiftbits) & mask], 1u);
  }
  __syncthreads();
  for (int i = threadIdx.x; i < nbins; i += blockDim.x)
    if (h[i]) atomicAdd(&hist[m * 2048 + i], h[i]);
}

__global__ void k_scan(const uint32_t* __restrict__ hist, uint32_t* __restrict__ state,
                       float* __restrict__ th, int nbits, int nbins, int final_pass) {
  int m = blockIdx.x;
  if (threadIdx.x != 0) return;
  uint32_t r = state[2*m+1];
  uint32_t prefix = state[2*m];
  uint32_t cum = 0;
  int sel = nbins - 1;
  for (int b = 0; b < nbins; b++) {
    uint32_t c = hist[m * 2048 + b];
    if (r < cum + c) { sel = b; break; }
    cum += c;
  }
  prefix = (prefix << nbits) | (uint32_t)sel;
  state[2*m]   = prefix;
  state[2*m+1] = r - cum;
  if (final_pass) th[m] = __uint_as_float(prefix);  // exact threshold value bits
}

// ---------------- row sums of filtered matrices (+1 for identity) ----------------

__global__ void k_rowsum(const float* __restrict__ attns, const float* __restrict__ th,
                         float* __restrict__ rs) {
  int m = blockIdx.y, row = blockIdx.x;
  float t = th[m];
  const float* base = attns + (size_t)(16 + m) * NNSQ + (size_t)row * NN;
  float s = 0.f;
  for (int k = threadIdx.x; k < NN; k += blockDim.x) {
    float x = base[k];
    s += (x > t) ? x : 0.f;
  }
  __shared__ float red[128];
  red[threadIdx.x] = s; __syncthreads();
  for (int o = 64; o > 0; o >>= 1) {
    if (threadIdx.x < o) red[threadIdx.x] += red[threadIdx.x + o];
    __syncthreads();
  }
  if (threadIdx.x == 0) rs[m * NN + row] = 1.0f + red[0];
}

// ---- P1 = selected rows of aug[11]; also write out s=3 and Q1 (pre-divided by rs10[col]) ----

__global__ void k_init(const float* __restrict__ attns, const int* __restrict__ pos,
                       const float* __restrict__ th, const float* __restrict__ rs,
                       float* __restrict__ Qa, float* __restrict__ out) {
  int idx = blockIdx.x * blockDim.x + threadIdx.x;
  if (idx >= 2 * 16 * QS) return;
  int k = idx % QS;
  int i = (idx / QS) % 16;
  int b = idx / (QS * 16);
  float* q = Qa + (size_t)(b * 16 + i) * QS + k;
  if (k >= NN) { *q = 0.f; return; }            // zero padding cols 1125..1127
  int m11 = 6 + b;                               // block 11
  int m10 = 4 + b;                               // block 10
  int r = 1025 + pos[b * 16 + i];
  float x = attns[(size_t)(16 + m11) * NNSQ + (size_t)r * NN + k];
  float f = (x > th[m11]) ? x : 0.f;
  if (k == r) f += 1.0f;                         // + identity
  float p = f / rs[m11 * NN + r];                // row-normalized aug[11] row
  *q = p / rs[m10 * NN + k];                     // Q for the next GEMM
  if (k >= 1 && k <= 1024)
    out[((size_t)(b * 16 + i) * 4 + 3) * 1024 + (k - 1)] = p;
}

// ---------------- chained (16 x 1125) @ (1125 x 1125) via V_WMMA_F32_16X16X4_F32 ----------------
// P_next[i,c] = sum_j Q[i,j]*f(a[j,c]) + Q[i,c];  Q_next[i,c] = P_next[i,c]/rs_next[c]

__global__ void __launch_bounds__(256) k_gemm(const float* __restrict__ attns,
                                              const float* __restrict__ th,
                                              const float* __restrict__ rs,
                                              const float* __restrict__ Qin,
                                              float* __restrict__ Qout,
                                              float* __restrict__ out,
                                              int blk, int s, int has_next) {
  int b    = blockIdx.y;
  int wave = threadIdx.x >> 5;
  int t    = blockIdx.x * 8 + wave;              // wave-uniform: EXEC stays full for WMMA
  if (t >= NTILES) return;
  int lane = threadIdx.x & 31;
  int half = lane >> 4, l16 = lane & 15;

  int m = (blk - 8) * 2 + b;
  float tv = th[m];
  const float* B    = attns + (size_t)(16 + m) * NNSQ;
  const float* Qb   = Qin + (size_t)b * 16 * QS;
  const float* Qrow = Qb + l16 * QS;

  int col  = t * 16 + l16;
  bool colOK = col < NN;
  int colC = colOK ? col : NN - 1;               // clamp: loads always in-bounds
  const float* Bc = B + colC;

  int ks0 = 2 * half;                            // A & B fragments: K = 2*half + vgpr
  v8f acc = {};
  // Full body: K = 0..1123  (ks+1 <= 1123 < 1125 -> branchless)
  #pragma unroll 4
  for (int k0 = 0; k0 < 1124; k0 += 4) {
    int ks = k0 + ks0;
    v2f a2 = *(const v2f*)(Qrow + ks);           // Q zero-padded -> always safe
    float x0 = Bc[(size_t)ks * NN];
    float x1 = Bc[(size_t)(ks + 1) * NN];
    v2f bv;
    bv.x = (x0 > tv) ? x0 : 0.f;
    bv.y = (x1 > tv) ? x1 : 0.f;
    acc = __builtin_amdgcn_wmma_f32_16x16x4_f32(false, a2, false, bv, (short)0, acc,
                                                false, false);
  }
  // Tail: K = 1124 only (half==1 lane group contributes zero, via cndmask not exec)
  {
    v2f a2 = *(const v2f*)(Qrow + 1124 + ks0);
    float x0 = Bc[(size_t)1124 * NN];
    v2f bv;
    bv.x = (half == 0 && x0 > tv) ? x0 : 0.f;
    bv.y = 0.f;
    acc = __builtin_amdgcn_wmma_f32_16x16x4_f32(false, a2, false, bv, (short)0, acc,
                                                false, false);
  }

  const float* rsn = has_next ? (rs + ((blk - 1 - 8) * 2 + b) * NN) : nullptr;
  float* Qo = has_next ? (Qout + (size_t)b * 16 * QS) : nullptr;
  #pragma unroll
  for (int v = 0; v < 8; v++) {
    int row = v + 8 * half;                      // C/D layout: M = vgpr + 8*half
    if (colOK) {
      float pv = acc[v] + Qb[row * QS + col];    // + folded identity term
      if (has_next) Qo[row * QS + col] = pv / rsn[col];
      if (col >= 1 && col <= 1024)
        out[((size_t)(b * 16 + row) * 4 + s) * 1024 + (col - 1)] = pv;
    } else if (has_next && col < QS) {
      Qo[row * QS + col] = 0.f;                  // keep padding zeroed
    }
  }
}

// ---------------- launcher ----------------

extern "C" void kernel_launch(void* const* d_in, const int* in_sizes, int n_in,
                              void* d_out, int out_size, void* d_ws, size_t ws_size,
                              hipStream_t stream) {
  const float* attns = (const float*)d_in[0];
  const int*   pos   = (const int*)d_in[1];
  float* out = (float*)d_out;

  char* ws = (char*)d_ws;
  uint32_t* hist  = (uint32_t*)(ws + 0);        // 8*2048*4   = 65536 B
  uint32_t* state = (uint32_t*)(ws + 65536);    // 8*uint2    = 64 B
  float*    th    = (float*)(ws + 65600);       // 8 floats
  float*    rs    = (float*)(ws + 65632);       // 8*1125 f32 = 36000 B
  float*    Qa    = (float*)(ws + 101632);      // 2*16*1128 f32 = 144384 B
  float*    Qb    = (float*)(ws + 246016);      // 144384 B

  // exact 10%-quantile thresholds for the 8 needed maps (blocks 8..11, both batches)
  k_zero<<<16, 256, 0, stream>>>(hist, state, 1);
  k_hist<<<dim3(64, 8), 256, 0, stream>>>(attns, hist, state, 0, 21, 2048);
  k_scan<<<8, 32, 0, stream>>>(hist, state, th, 11, 2048, 0);
  k_zero<<<16, 256, 0, stream>>>(hist, state, 0);
  k_hist<<<dim3(64, 8), 256, 0, stream>>>(attns, hist, state, 11, 10, 2048);
  k_scan<<<8, 32, 0, stream>>>(hist, state, th, 11, 2048, 0);
  k_zero<<<16, 256, 0, stream>>>(hist, state, 0);
  k_hist<<<dim3(64, 8), 256, 0, stream>>>(attns, hist, state, 22, 0, 1024);
  k_scan<<<8, 32, 0, stream>>>(hist, state, th, 10, 1024, 1);

  k_rowsum<<<dim3(1125, 8), 128, 0, stream>>>(attns, th, rs);

  k_init<<<(2 * 16 * QS + 255) / 256, 256, 0, stream>>>(attns, pos, th, rs, Qa, out);

  k_gemm<<<dim3(9, 2), 256, 0, stream>>>(attns, th, rs, Qa, Qb, out, 10, 2, 1);
  k_gemm<<<dim3(9, 2), 256, 0, stream>>>(attns, th, rs, Qb, Qa, out,  9, 1, 1);
  k_gemm<<<dim3(9, 2), 256, 0, stream>>>(attns, th, rs, Qa, nullptr, out, 8, 0, 0);
}